// CFDErrorInterpolate_28767690948643
// MI455X (gfx1250) — compile-verified
//
#include <hip/hip_runtime.h>
#include <hip/hip_bf16.h>

// ---------------------------------------------------------------------------
// CDNA5 (gfx1250) implementation of CFDError + knn-interpolate + GCN enc/dec.
// Dense GEMMs use V_WMMA_F32_16X16X4_F32 (true fp32 matrix op, wave32).
// Graph aggregation uses native f32 global atomics (bandwidth-bound part).
// ---------------------------------------------------------------------------

typedef float v2f __attribute__((ext_vector_type(2)));
typedef float v8f __attribute__((ext_vector_type(8)));

#define NL 8192
#define NH 32768
#define EL 65536
#define EH 262144

static __device__ __forceinline__ v8f wmma_f32(v2f a, v2f b, v8f c) {
  return __builtin_amdgcn_wmma_f32_16x16x4_f32(false, a, false, b, (short)0, c,
                                               false, false);
}

// atomic max for non-negative floats via monotonic int-bit trick
static __device__ __forceinline__ void atomic_max_pos(float* addr, float v) {
  atomicMax((int*)addr, __float_as_int(v));
}

// ---------------------------------------------------------------------------
__global__ void fill_kernel(float* p, float v, int n) {
  int i = blockIdx.x * blockDim.x + threadIdx.x;
  if (i < n) p[i] = v;
}

__global__ void add3_kernel(const float* a, const float* b, const float* c,
                            float* o, int n) {
  int i = blockIdx.x * blockDim.x + threadIdx.x;
  if (i < n) o[i] = a[i] + b[i] + c[i];
}

// ---- EdgeConv #0: in=2 (pos), concat 4, out 64, scalar ---------------------
__global__ void ec_init_kernel(const float* __restrict__ pos,
                               const int* __restrict__ src,
                               const int* __restrict__ dst,
                               const float* __restrict__ W,  // [4,64]
                               const float* __restrict__ b,
                               float* __restrict__ out,      // [NL,64] zeroed
                               int E) {
  int e = blockIdx.x * blockDim.x + threadIdx.x;
  if (e >= E) return;
  int s = src[e], d = dst[e];
  float f0 = pos[2 * d], f1 = pos[2 * d + 1];
  float f2 = pos[2 * s] - f0, f3 = pos[2 * s + 1] - f1;
  float* op = out + (size_t)d * 64;
  for (int c = 0; c < 64; ++c) {
    float m = b[c] + f0 * W[c] + f1 * W[64 + c] + f2 * W[128 + c] + f3 * W[192 + c];
    atomic_max_pos(op + c, fmaxf(m, 0.0f));
  }
}

// ---- pack ec1/ec2/ec4a/ec4b weights into one [128,256] matrix --------------
__global__ void build_wcat_kernel(const float* w1, const float* w2,
                                  const float* w3, const float* w4,
                                  const float* b1, const float* b2,
                                  const float* b3, const float* b4,
                                  float* Wcat, float* bcat) {
  int idx = blockIdx.x * blockDim.x + threadIdx.x;
  if (idx < 128 * 256) {
    int k = idx >> 8, c = idx & 255;
    int j = c >> 6, cc = c & 63;
    const float* w = (j == 0) ? w1 : (j == 1) ? w2 : (j == 2) ? w3 : w4;
    Wcat[idx] = w[k * 64 + cc];
  } else if (idx < 128 * 256 + 256) {
    int c = idx - 128 * 256;
    int j = c >> 6, cc = c & 63;
    const float* bb = (j == 0) ? b1 : (j == 1) ? b2 : (j == 2) ? b3 : b4;
    bcat[c] = bb[cc];
  }
}

// ---- fused EdgeConv GEMM: A=[xi|xj-xi] (K=128) x Wcat[128,256], relu,
// ---- atomic-max scatter into H4[NL,256]. 32 edges x 64 cols per block. -----
__global__ __launch_bounds__(128) void edge_conv_wmma_kernel(
    const float* __restrict__ h,   // [NL,64]
    const int* __restrict__ src, const int* __restrict__ dst,
    const float* __restrict__ Wcat,  // [128,256]
    const float* __restrict__ bcat,  // [256]
    float* __restrict__ out,         // [NL,256] zeroed
    int E) {
  __shared__ float Alds[32 * 132];
  __shared__ float Wlds[64 * 132];
  const int tid = threadIdx.x;
  const int e0 = blockIdx.x * 32;
  const int n0 = blockIdx.y * 64;
  for (int idx = tid; idx < 64 * 128; idx += 128) {
    int n = idx >> 7, k = idx & 127;
    Wlds[n * 132 + k] = Wcat[k * 256 + n0 + n];  // transposed: Wlds[n][k]
  }
  for (int idx = tid; idx < 32 * 128; idx += 128) {
    int m = idx >> 7, k = idx & 127;
    int e = e0 + m;
    int s = src[e], d = dst[e];
    float xi = h[(size_t)d * 64 + (k & 63)];
    Alds[m * 132 + k] = (k < 64) ? xi : (h[(size_t)s * 64 + (k & 63)] - xi);
  }
  __syncthreads();
  const int lane = tid & 31, w = tid >> 5;
  const int half = lane >> 4;
  const int hk = half << 1, nl = lane & 15;
  v8f acc0 = {}, acc1 = {};
  const float* wp = &Wlds[(w * 16 + nl) * 132 + hk];
  const float* ap = &Alds[(lane & 15) * 132 + hk];
  for (int kk = 0; kk < 32; ++kk) {
    v2f b = *(const v2f*)(wp + 4 * kk);
    v2f a0 = *(const v2f*)(ap + 4 * kk);
    v2f a1 = *(const v2f*)(ap + 16 * 132 + 4 * kk);
    acc0 = wmma_f32(a0, b, acc0);
    acc1 = wmma_f32(a1, b, acc1);
  }
  const int gc = n0 + w * 16 + nl;
  const float bb = bcat[gc];
#pragma unroll
  for (int v = 0; v < 8; ++v) {
    int m = v + (half << 3);
    int d0 = dst[e0 + m];
    atomic_max_pos(out + (size_t)d0 * 256 + gc, fmaxf(acc0[v] + bb, 0.0f));
    int d1 = dst[e0 + 16 + m];
    atomic_max_pos(out + (size_t)d1 * 256 + gc, fmaxf(acc1[v] + bb, 0.0f));
  }
}

// ---- cat = [x1, x2^2, ta^2+tb^2] ------------------------------------------
__global__ void combine_cat_kernel(const float* __restrict__ H4,
                                   float* __restrict__ cat) {
  int i = blockIdx.x * blockDim.x + threadIdx.x;
  if (i >= NL * 64) return;
  int n = i >> 6, c = i & 63;
  const float* r = H4 + (size_t)n * 256;
  float x1 = r[c], x2 = r[64 + c], ta = r[128 + c], tb = r[192 + c];
  float* o = cat + (size_t)n * 192;
  o[c] = x1;
  o[64 + c] = x2 * x2;
  o[128 + c] = ta * ta + tb * tb;
}

// ---- EdgeConv out: in=192, concat 384, out 3, scalar -----------------------
__global__ __launch_bounds__(256) void ec_out_kernel(
    const float* __restrict__ cat, const int* __restrict__ src,
    const int* __restrict__ dst, const float* __restrict__ W,  // [384,3]
    const float* __restrict__ b, float* __restrict__ out, int E) {
  __shared__ float sW[384 * 3];
  for (int t = threadIdx.x; t < 384 * 3; t += 256) sW[t] = W[t];
  __syncthreads();
  int e = blockIdx.x * blockDim.x + threadIdx.x;
  if (e >= E) return;
  int s = src[e], d = dst[e];
  const float* cd = cat + (size_t)d * 192;
  const float* cs = cat + (size_t)s * 192;
  float a0 = b[0], a1 = b[1], a2 = b[2];
  for (int k = 0; k < 192; ++k) {
    float xi = cd[k];
    float xj = cs[k] - xi;
    const float* w0 = sW + k * 3;
    const float* w1 = sW + (192 + k) * 3;
    a0 += xi * w0[0] + xj * w1[0];
    a1 += xi * w0[1] + xj * w1[1];
    a2 += xi * w0[2] + xj * w1[2];
  }
  float* op = out + (size_t)d * 3;
  atomic_max_pos(op + 0, fmaxf(a0, 0.0f));
  atomic_max_pos(op + 1, fmaxf(a1, 0.0f));
  atomic_max_pos(op + 2, fmaxf(a2, 0.0f));
}

__global__ void build_feat_kernel(const float* __restrict__ x_l,
                                  const float* __restrict__ e,
                                  float* __restrict__ feat) {
  int i = blockIdx.x * blockDim.x + threadIdx.x;
  if (i >= NL) return;
  float* f = feat + (size_t)i * 6;
  f[0] = x_l[3 * i]; f[1] = x_l[3 * i + 1]; f[2] = x_l[3 * i + 2];
  f[3] = e[3 * i];   f[4] = e[3 * i + 1];   f[5] = e[3 * i + 2];
}

// ---- brute-force 32-NN + inverse-distance interpolate ----------------------
__global__ __launch_bounds__(256) void knn_kernel(
    const float* __restrict__ pos_l, const float* __restrict__ pos_h,
    const float* __restrict__ feat, float* __restrict__ xh) {
  __shared__ float sx[2048], sy[2048];
  int i = blockIdx.x * 256 + threadIdx.x;  // grid covers exactly NH
  float px = pos_h[2 * i], py = pos_h[2 * i + 1];
  float dk[32];
  int ik[32];
#pragma unroll
  for (int k = 0; k < 32; ++k) { dk[k] = 3.4e38f; ik[k] = 0; }
  for (int c0 = 0; c0 < NL; c0 += 2048) {
    __syncthreads();
    for (int t = threadIdx.x; t < 2048; t += 256) {
      sx[t] = pos_l[2 * (c0 + t)];
      sy[t] = pos_l[2 * (c0 + t) + 1];
      if (c0 + 2048 < NL)  // warm caches for next tile (global_prefetch_b8)
        __builtin_prefetch(pos_l + 2 * (c0 + 2048 + t), 0, 1);
    }
    __syncthreads();
    for (int j = 0; j < 2048; ++j) {
      float dx = px - sx[j], dy = py - sy[j];
      float dd = dx * dx + dy * dy;
      if (dd < dk[31]) {
        dk[31] = dd; ik[31] = c0 + j;
#pragma unroll
        for (int k = 31; k > 0; --k) {
          if (dk[k] < dk[k - 1]) {
            float tv = dk[k]; dk[k] = dk[k - 1]; dk[k - 1] = tv;
            int ti = ik[k]; ik[k] = ik[k - 1]; ik[k - 1] = ti;
          }
        }
      }
    }
  }
  float wsum = 0.0f;
  float acc[6] = {0, 0, 0, 0, 0, 0};
#pragma unroll
  for (int k = 0; k < 32; ++k) {
    float w = 1.0f / fmaxf(dk[k], 1e-16f);
    wsum += w;
    const float* f = feat + (size_t)ik[k] * 6;
#pragma unroll
    for (int c = 0; c < 6; ++c) acc[c] += w * f[c];
  }
  float inv = 1.0f / wsum;
  float* o = xh + (size_t)i * 6;
#pragma unroll
  for (int c = 0; c < 6; ++c) o[c] = acc[c] * inv;
}

// ---- degree / dinv ---------------------------------------------------------
__global__ void deg_kernel(float* deg, const int* __restrict__ dst, int E) {
  int e = blockIdx.x * blockDim.x + threadIdx.x;
  if (e < E) atomicAdd(&deg[dst[e]], 1.0f);
}
__global__ void rsqrt_kernel(float* d, int n) {
  int i = blockIdx.x * blockDim.x + threadIdx.x;
  if (i < n) d[i] = rsqrtf(fmaxf(d[i], 1e-12f));
}

// ---- generic fp32 WMMA GEMM ------------------------------------------------
// C[M,N] = (accum ? C : 0) + A[M,K] @ W[K,N] + bias
// If H != null, additionally H = C*dinv^2 + bias2 (fused GCN self-loop term).
// block = 128 threads (4 waves), tile = 32 rows x 64 cols.
__global__ __launch_bounds__(128) void gemm_wmma_kernel(
    const float* __restrict__ A, const float* __restrict__ W,
    const float* __restrict__ bias, float* __restrict__ C,
    float* __restrict__ H, const float* __restrict__ dinv,
    const float* __restrict__ bias2, int M, int K, int N, int accum) {
  __shared__ float Alds[32 * 132];
  __shared__ float Wlds[64 * 132];
  const int tid = threadIdx.x;
  const int row0 = blockIdx.x * 32;
  const int n0 = blockIdx.y * 64;
  const int KP = (K + 3) & ~3;
  for (int idx = tid; idx < 64 * KP; idx += 128) {
    int n = idx / KP, k = idx - n * KP;
    int gn = n0 + n;
    Wlds[n * 132 + k] = (k < K && gn < N) ? W[(size_t)k * N + gn] : 0.0f;
  }
  for (int idx = tid; idx < 32 * KP; idx += 128) {
    int m = idx / KP, k = idx - m * KP;
    int r = row0 + m;
    Alds[m * 132 + k] = (k < K && r < M) ? A[(size_t)r * K + k] : 0.0f;
  }
  __syncthreads();
  const int lane = tid & 31, w = tid >> 5;
  const int half = lane >> 4;
  const int hk = half << 1, nl = lane & 15;
  v8f acc0 = {}, acc1 = {};
  const float* wp = &Wlds[(w * 16 + nl) * 132 + hk];
  const float* ap = &Alds[(lane & 15) * 132 + hk];
  const int K4 = KP >> 2;
  for (int kk = 0; kk < K4; ++kk) {
    v2f b = *(const v2f*)(wp + 4 * kk);
    v2f a0 = *(const v2f*)(ap + 4 * kk);
    v2f a1 = *(const v2f*)(ap + 16 * 132 + 4 * kk);
    acc0 = wmma_f32(a0, b, acc0);
    acc1 = wmma_f32(a1, b, acc1);
  }
  const int gn = n0 + w * 16 + nl;
  if (gn < N) {
    const float bb = bias ? bias[gn] : 0.0f;
    const float b2 = bias2 ? bias2[gn] : 0.0f;
#pragma unroll
    for (int v = 0; v < 8; ++v) {
      int r0 = row0 + v + (half << 3);
      int r1 = r0 + 16;
      if (r0 < M) {
        float val = acc0[v] + bb;
        if (accum) val += C[(size_t)r0 * N + gn];
        C[(size_t)r0 * N + gn] = val;
        if (H) { float di = dinv[r0]; H[(size_t)r0 * N + gn] = val * di * di + b2; }
      }
      if (r1 < M) {
        float val = acc1[v] + bb;
        if (accum) val += C[(size_t)r1 * N + gn];
        C[(size_t)r1 * N + gn] = val;
        if (H) { float di = dinv[r1]; H[(size_t)r1 * N + gn] = val * di * di + b2; }
      }
    }
  }
}

// ---- GCN self-loop + bias (standalone, for the fused-m path) ---------------
__global__ void self_bias_kernel(float* __restrict__ out,
                                 const float* __restrict__ xw,
                                 const float* __restrict__ dinv,
                                 const float* __restrict__ b, int N, int C) {
  int i = blockIdx.x * blockDim.x + threadIdx.x;
  if (i >= N * C) return;
  int n = i / C, c = i - n * C;
  float di = dinv[n];
  out[i] = xw[i] * di * di + b[c];
}

// ---- GCN edge scatter-add, C=128 (wave per edge, float4 per lane) ----------
__global__ __launch_bounds__(256) void edge_scatter128_kernel(
    float* __restrict__ out, const float* __restrict__ xw,
    const float* __restrict__ dinv, const int* __restrict__ src,
    const int* __restrict__ dst, int E) {
  int lane = threadIdx.x & 31;
  int e = (blockIdx.x * blockDim.x + threadIdx.x) >> 5;
  if (e >= E) return;
  int s = src[e], d = dst[e];
  float coef = dinv[s] * dinv[d];
  const float4 xv = *(const float4*)(xw + (size_t)s * 128 + lane * 4);
  float* op = out + (size_t)d * 128 + lane * 4;
  atomicAdd(op + 0, xv.x * coef);
  atomicAdd(op + 1, xv.y * coef);
  atomicAdd(op + 2, xv.z * coef);
  atomicAdd(op + 3, xv.w * coef);
}

// ---- GCN edge scatter-add, C=3 ---------------------------------------------
__global__ void edge_scatter3_kernel(float* __restrict__ out,
                                     const float* __restrict__ xw,
                                     const float* __restrict__ dinv,
                                     const int* __restrict__ src,
                                     const int* __restrict__ dst, int E) {
  int e = blockIdx.x * blockDim.x + threadIdx.x;
  if (e >= E) return;
  int s = src[e], d = dst[e];
  float coef = dinv[s] * dinv[d];
  atomicAdd(out + (size_t)d * 3 + 0, xw[(size_t)s * 3 + 0] * coef);
  atomicAdd(out + (size_t)d * 3 + 1, xw[(size_t)s * 3 + 1] * coef);
  atomicAdd(out + (size_t)d * 3 + 2, xw[(size_t)s * 3 + 2] * coef);
}

// ---- BatchNorm statistics --------------------------------------------------
__global__ __launch_bounds__(256) void bn_partial_kernel(
    const float* __restrict__ h, float* __restrict__ asum,
    float* __restrict__ asq, int N, int C) {
  __shared__ float ls[128], lq[128];
  for (int t = threadIdx.x; t < C; t += 256) { ls[t] = 0.0f; lq[t] = 0.0f; }
  __syncthreads();
  int total = N * C;
  for (int i = blockIdx.x * blockDim.x + threadIdx.x; i < total;
       i += gridDim.x * blockDim.x) {
    float v = h[i];
    int c = i % C;
    atomicAdd(&ls[c], v);
    atomicAdd(&lq[c], v * v);
  }
  __syncthreads();
  for (int t = threadIdx.x; t < C; t += 256) {
    atomicAdd(&asum[t], ls[t]);
    atomicAdd(&asq[t], lq[t]);
  }
}
__global__ void bn_final_kernel(const float* asum, const float* asq, float* mu,
                                float* rstd, int N, int C) {
  int c = threadIdx.x;
  if (c >= C) return;
  float m = asum[c] / (float)N;
  float var = asq[c] / (float)N - m * m;
  mu[c] = m;
  rstd[c] = rsqrtf(var + 1e-5f);
}
__global__ void bn_relu_res_kernel(float* __restrict__ out,
                                   const float* __restrict__ h,
                                   const float* __restrict__ mu,
                                   const float* __restrict__ rstd,
                                   const float* __restrict__ g,
                                   const float* __restrict__ beta,
                                   const float* __restrict__ ident, int N,
                                   int C) {
  int i = blockIdx.x * blockDim.x + threadIdx.x;
  if (i >= N * C) return;
  int c = i % C;
  float v = (h[i] - mu[c]) * rstd[c] * g[c] + beta[c];
  out[i] = fmaxf(v, 0.0f) + ident[i];
}

// ---------------------------------------------------------------------------
extern "C" void kernel_launch(void* const* d_in, const int* in_sizes, int n_in,
                              void* d_out, int out_size, void* d_ws,
                              size_t ws_size, hipStream_t stream) {
  (void)n_in; (void)out_size; (void)ws_size;
  auto F = [&](int i) { return (const float*)d_in[i]; };
  auto I = [&](int i) { return (const int*)d_in[i]; };
  const float *x_l, *pos_l, *pos_h;
  const int *eil, *eih;
  const float *ecinitW, *ecinitB, *ec1W, *ec1B, *ec2W, *ec2B, *ec4aW, *ec4aB,
      *ec4bW, *ec4bB, *ecoutW, *ecoutB;
  const float *e1gW, *e1gB, *e1g, *e1be, *e1rW, *e1rB;
  const float *e2gW, *e2gB, *e2g, *e2be;
  const float *e3gW, *e3gB, *e3g, *e3be;
  const float *m1W, *m1B, *m2W, *m2B, *m3W, *m3B;
  const float *d1gW, *d1gB, *d1g, *d1be;
  const float *d2gW, *d2gB, *d2g, *d2be, *d2rW, *d2rB;
  const float *d3gW, *d3gB, *d3g, *d3be;
  if (in_sizes[0] == NL * 3) {  // dict insertion-order flattening
    x_l = F(0); pos_l = F(1); pos_h = F(2); eil = I(3); eih = I(4);
    ecinitW = F(5); ecinitB = F(6);
    ec1W = F(7); ec1B = F(8); ec2W = F(9); ec2B = F(10);
    ec4aW = F(11); ec4aB = F(12); ec4bW = F(13); ec4bB = F(14);
    ecoutW = F(15); ecoutB = F(16);
    e1gW = F(17); e1gB = F(18); e1g = F(19); e1be = F(20); e1rW = F(21); e1rB = F(22);
    e2gW = F(23); e2gB = F(24); e2g = F(25); e2be = F(26);
    e3gW = F(27); e3gB = F(28); e3g = F(29); e3be = F(30);
    m1W = F(31); m1B = F(32); m2W = F(33); m2B = F(34); m3W = F(35); m3B = F(36);
    d1gW = F(37); d1gB = F(38); d1g = F(39); d1be = F(40);
    d2gW = F(41); d2gB = F(42); d2g = F(43); d2be = F(44); d2rW = F(45); d2rB = F(46);
    d3gW = F(47); d3gB = F(48); d3g = F(49); d3be = F(50);
  } else {  // jax tree_flatten sorted-key order
    eih = I(0); eil = I(1);
    d1be = F(2); d1g = F(3); d1gW = F(4); d1gB = F(5);
    d2be = F(6); d2g = F(7); d2gW = F(8); d2gB = F(9); d2rW = F(10); d2rB = F(11);
    d3be = F(12); d3g = F(13); d3gW = F(14); d3gB = F(15);
    ec1W = F(16); ec1B = F(17); ec2W = F(18); ec2B = F(19);
    ec4aW = F(20); ec4aB = F(21); ec4bW = F(22); ec4bB = F(23);
    ecinitW = F(24); ecinitB = F(25); ecoutW = F(26); ecoutB = F(27);
    e1be = F(28); e1g = F(29); e1gW = F(30); e1gB = F(31); e1rW = F(32); e1rB = F(33);
    e2be = F(34); e2g = F(35); e2gW = F(36); e2gB = F(37);
    e3be = F(38); e3g = F(39); e3gW = F(40); e3gB = F(41);
    m1W = F(42); m1B = F(43); m2W = F(44); m2B = F(45); m3W = F(46); m3B = F(47);
    pos_h = F(48); pos_l = F(49); x_l = F(50);
  }
  const int *sl = eil, *dl = eil + EL, *sh = eih, *dh = eih + EH;

  // ----- workspace layout (floats) -----
  float* P = (float*)d_ws;
  const size_t BIG = (size_t)NH * 128;  // 4,194,304
  float* XH = P;                        // [NH,6]
  float* B0 = P + 196608;
  float* B1 = B0 + BIG;
  float* B2 = B1 + BIG;
  float* B3 = B2 + BIG;
  float* B4 = B3 + BIG;
  float* B5 = B4 + BIG;
  float* SM = B5 + BIG;
  float* dinv = SM;               // 32768
  float* bnsum = SM + 32768;      // 128
  float* bnsq = bnsum + 128;      // 128
  float* mu = bnsq + 128;         // 128
  float* rstd = mu + 128;         // 128
  float* bsum = rstd + 128;       // 128
  float* id3 = bsum + 128;        // [NH,3]
  float* g3a = id3 + NH * 3;      // [NH,3]
  float* g3b = g3a + NH * 3;      // [NH,3]
  float* d2o = g3b + NH * 3;      // [NH,3]
  float* d3xw = d2o + NH * 3;     // [NH,3]
  float* d3h = d3xw + NH * 3;     // [NH,3]
  // low-phase aliases (dead before high phase uses B0..B5)
  float* h64 = B2;                // [NL,64]
  float* H4 = B3;                 // [NL,256]
  float* catb = B4;               // [NL,192]
  float* Eo = B5;                 // [NL,3]
  float* feat = B5 + NL * 3;      // [NL,6]
  float* Wcat = B0;               // [128,256]
  float* bcat = B0 + 128 * 256;   // [256]

  auto fill = [&](float* p, float v, int n) {
    fill_kernel<<<dim3((n + 255) / 256), dim3(256), 0, stream>>>(p, v, n);
  };
  // plain GEMM (optionally accumulate into C)
  auto gemm = [&](const float* A, const float* W, const float* bias, float* C,
                  int M, int K, int N, int accum) {
    dim3 g(M / 32, (N + 63) / 64);
    gemm_wmma_kernel<<<g, dim3(128), 0, stream>>>(
        A, W, bias, C, nullptr, nullptr, nullptr, M, K, N, accum);
  };
  // GEMM with fused GCN self-loop term: C = A@W, H = C*dinv^2 + b2
  auto gemm_self = [&](const float* A, const float* W, float* C, float* H,
                       const float* b2, int K, int N) {
    dim3 g(NH / 32, (N + 63) / 64);
    gemm_wmma_kernel<<<g, dim3(128), 0, stream>>>(
        A, W, nullptr, C, H, dinv, b2, NH, K, N, 0);
  };
  auto scat128 = [&](float* out, const float* xw) {
    edge_scatter128_kernel<<<dim3(EH * 32 / 256), dim3(256), 0, stream>>>(
        out, xw, dinv, sh, dh, EH);
  };
  auto scat3 = [&](float* out, const float* xw) {
    edge_scatter3_kernel<<<dim3((EH + 255) / 256), dim3(256), 0, stream>>>(
        out, xw, dinv, sh, dh, EH);
  };
  auto bn_stats = [&](const float* hbuf, int C) {
    fill(bnsum, 0.0f, C);
    fill(bnsq, 0.0f, C);
    bn_partial_kernel<<<dim3(1024), dim3(256), 0, stream>>>(hbuf, bnsum, bnsq,
                                                            NH, C);
    bn_final_kernel<<<dim3(1), dim3(128), 0, stream>>>(bnsum, bnsq, mu, rstd,
                                                       NH, C);
  };
  auto bn_relu_res = [&](float* out, const float* hbuf, const float* g,
                         const float* be, const float* ident, int C) {
    bn_relu_res_kernel<<<dim3((NH * C + 255) / 256), dim3(256), 0, stream>>>(
        out, hbuf, mu, rstd, g, be, ident, NH, C);
  };

  // ================= low-res: CFDError =================
  fill(h64, 0.0f, NL * 64);
  ec_init_kernel<<<dim3(EL / 256), dim3(256), 0, stream>>>(
      pos_l, sl, dl, ecinitW, ecinitB, h64, EL);
  build_wcat_kernel<<<dim3((128 * 256 + 256 + 255) / 256), dim3(256), 0,
                      stream>>>(ec1W, ec2W, ec4aW, ec4bW, ec1B, ec2B, ec4aB,
                                ec4bB, Wcat, bcat);
  fill(H4, 0.0f, NL * 256);
  edge_conv_wmma_kernel<<<dim3(EL / 32, 4), dim3(128), 0, stream>>>(
      h64, sl, dl, Wcat, bcat, H4, EL);
  combine_cat_kernel<<<dim3(NL * 64 / 256), dim3(256), 0, stream>>>(H4, catb);
  fill(Eo, 0.0f, NL * 3);
  ec_out_kernel<<<dim3(EL / 256), dim3(256), 0, stream>>>(catb, sl, dl, ecoutW,
                                                          ecoutB, Eo, EL);
  build_feat_kernel<<<dim3(NL / 256), dim3(256), 0, stream>>>(x_l, Eo, feat);

  // ================= knn interpolate =================
  knn_kernel<<<dim3(NH / 256), dim3(256), 0, stream>>>(pos_l, pos_h, feat, XH);

  // ================= degrees (high graph, incl. self loop) =================
  fill(dinv, 1.0f, NH);
  deg_kernel<<<dim3((EH + 255) / 256), dim3(256), 0, stream>>>(dinv, dh, EH);
  rsqrt_kernel<<<dim3(NH / 256), dim3(256), 0, stream>>>(dinv, NH);

  // ================= encoder / decoder =================
  // enc1 (6 -> 128, with residual projection)
  gemm_self(XH, e1gW, B0, B1, e1gB, 6, 128);
  scat128(B1, B0);
  bn_stats(B1, 128);
  gemm(XH, e1rW, e1rB, B0, NH, 6, 128, 0);   // ident
  bn_relu_res(B2, B1, e1g, e1be, B0, 128);   // E1 = B2
  // enc2
  gemm_self(B2, e2gW, B0, B1, e2gB, 128, 128);
  scat128(B1, B0);
  bn_stats(B1, 128);
  bn_relu_res(B3, B1, e2g, e2be, B2, 128);   // E2 = B3
  // enc3
  gemm_self(B3, e3gW, B0, B1, e3gB, 128, 128);
  scat128(B1, B0);
  bn_stats(B1, 128);
  bn_relu_res(B4, B1, e3g, e3be, B3, 128);   // E3 = B4
  // m = gcn(E1,mp1)+gcn(E2,mp2)+gcn(E3,mp3): aggregation is linear in xw,
  // so accumulate xw = E1@W1 + E2@W2 + E3@W3 and do ONE edge scatter pass.
  gemm(B2, m1W, nullptr, B0, NH, 128, 128, 0);
  gemm(B3, m2W, nullptr, B0, NH, 128, 128, 1);
  gemm(B4, m3W, nullptr, B0, NH, 128, 128, 1);
  add3_kernel<<<dim3(1), dim3(128), 0, stream>>>(m1B, m2B, m3B, bsum, 128);
  self_bias_kernel<<<dim3(NH * 128 / 256), dim3(256), 0, stream>>>(
      B5, B0, dinv, bsum, NH, 128);
  scat128(B5, B0);                           // M = B5
  // dec1 (128 -> 128)
  gemm_self(B5, d1gW, B0, B1, d1gB, 128, 128);
  scat128(B1, B0);
  bn_stats(B1, 128);
  bn_relu_res(B2, B1, d1g, d1be, B5, 128);   // D1 = B2
  // dec2 (128 -> 3, with residual projection)
  gemm_self(B2, d2gW, g3a, g3b, d2gB, 128, 3);
  scat3(g3b, g3a);
  bn_stats(g3b, 3);
  gemm(B2, d2rW, d2rB, id3, NH, 128, 3, 0);
  bn_relu_res(d2o, g3b, d2g, d2be, id3, 3);
  // dec3 (3 -> 3)
  gemm_self(d2o, d3gW, d3xw, d3h, d3gB, 3, 3);
  scat3(d3h, d3xw);
  bn_stats(d3h, 3);
  bn_relu_res((float*)d_out, d3h, d3g, d3be, d2o, 3);
}